// PathDTWBatchTF_31568009625646
// MI455X (gfx1250) — compile-verified
//
#include <hip/hip_runtime.h>

// Problem constants from the reference.
constexpr int   BATCH = 64;
constexpr int   M = 256;
constexpr int   N = 256;
constexpr float GAMMA = 0.01f;
constexpr float INVG  = 1.0f / GAMMA;     // 100
constexpr float BIGV  = 1.0e10f;

typedef float v2f __attribute__((ext_vector_type(2)));
typedef float v8f __attribute__((ext_vector_type(8)));

// ---------------------------------------------------------------------------
// One workgroup (256 threads = 8 wave32) per batch matrix.
// Forward soft-DTW DP over anti-diagonals with a 3-deep LDS ring of V
// diagonals; stores R = softmin value (V - theta) to workspace.
// Backward E-recursion over anti-diagonals (reverse order) with 3-deep LDS
// rings for E and R; weights recomputed as exp((R_neighbor - V)/gamma).
// ---------------------------------------------------------------------------
__global__ void __launch_bounds__(256)
dtw_grad_kernel(const float* __restrict__ theta_all,
                float* __restrict__ R_ws,
                float* __restrict__ E_ws) {
    const int b = blockIdx.x;
    const int t = threadIdx.x;                 // 0..255
    const float* th = theta_all + (size_t)b * M * N;
    float* Rb = R_ws + (size_t)b * M * N;
    float* Eb = E_ws + (size_t)b * M * N;

    // ---- forward: V diagonals ring, indexed by i (0..M), diag d = i+j ----
    __shared__ float vd[3][M + 1];

    if (t == 0) {
        vd[0][0] = 0.0f;          // V[0][0]
        vd[1][0] = BIGV;          // V[0][1]
        vd[1][1] = BIGV;          // V[1][0]
    }
    __syncthreads();

    for (int d = 2; d <= M + N; ++d) {
        float*       cur = vd[d % 3];
        const float* p1  = vd[(d - 1) % 3];    // diag d-1
        const float* p2  = vd[(d - 2) % 3];    // diag d-2
        const int ilo = (d - N > 1) ? (d - N) : 1;
        const int ihi = (d - 1 < M) ? (d - 1) : M;
        const int i = ilo + t;
        if (i <= ihi) {
            const int j = d - i;
            const float a  = p1[i];            // V[i][j-1]
            const float dg = p2[i - 1];        // V[i-1][j-1]
            const float c  = p1[i - 1];        // V[i-1][j]
            float mn = fminf(a, fminf(dg, c));
            float s  = __expf((mn - a) * INVG)
                     + __expf((mn - dg) * INVG)
                     + __expf((mn - c) * INVG);
            float r  = mn - GAMMA * __logf(s);               // softmin
            cur[i] = th[(i - 1) * N + (j - 1)] + r;          // V[i][j]
            Rb[(i - 1) * N + (j - 1)] = r;
        }
        if (t == 0 && d <= N) cur[0] = BIGV;   // V[0][d]
        if (t == 1 && d <= M) cur[d] = BIGV;   // V[d][0]
        __syncthreads();
    }

    // ---- backward: E and R diagonal rings, slots indexed by i (0..M+1) ----
    __shared__ float ed[3][M + 2];
    __shared__ float rd[3][M + 2];
    for (int k = 0; k < 3; ++k) {
        ed[k][t] = 0.0f;  rd[k][t] = -BIGV;
        if (t == 0) {
            ed[k][M]     = 0.0f;  rd[k][M]     = -BIGV;
            ed[k][M + 1] = 0.0f;  rd[k][M + 1] = -BIGV;
        }
    }
    __syncthreads();

    for (int d = M + N; d >= 2; --d) {
        float*       curE = ed[d % 3];
        float*       curR = rd[d % 3];
        const float* e1 = ed[(d + 1) % 3];     // diag d+1
        const float* r1 = rd[(d + 1) % 3];
        const float* e2 = ed[(d + 2) % 3];     // diag d+2
        const float* r2 = rd[(d + 2) % 3];
        const int ilo = (d - N > 1) ? (d - N) : 1;
        const int ihi = (d - 1 < M) ? (d - 1) : M;
        const int i = t + 1;                   // this thread owns slot i
        float ev = 0.0f, rv = -BIGV;
        if (i >= ilo && i <= ihi) {
            const int j = d - i;
            const float r_ij = Rb[(i - 1) * N + (j - 1)];
            const float v_ij = r_ij + th[(i - 1) * N + (j - 1)];
            // Q[i,j+1,0]*E[i,j+1] + Q[i+1,j+1,1]*E[i+1,j+1] + Q[i+1,j,2]*E[i+1,j]
            float e = __expf((r1[i]     - v_ij) * INVG) * e1[i]       // right
                    + __expf((r2[i + 1] - v_ij) * INVG) * e2[i + 1]   // diag
                    + __expf((r1[i + 1] - v_ij) * INVG) * e1[i + 1];  // down
            if (i == M && j == N) e += 1.0f;   // Q[M+1][N+1]=1, E[M+1][N+1]=1
            ev = e; rv = r_ij;
            Eb[(i - 1) * N + (j - 1)] = e;
        }
        curE[i] = ev;                          // defaults outside range
        curR[i] = rv;
        if (t == 0) { curE[M + 1] = 0.0f; curR[M + 1] = -BIGV; }
        __syncthreads();
    }
}

// ---------------------------------------------------------------------------
// Deterministic batch mean via V_WMMA_F32_16X16X4_F32 (full fp32).
// Per wave: 16 output elements. A[16x4] holds E values for 4 batches
// (contraction over batch), B = 1/64 everywhere, accumulate 16 WMMAs to
// cover all 64 batches. D[r][c] is independent of c; lanes 0 and 16 hold
// D rows M=0..7 / M=8..15 in their 8 accumulator VGPRs and write them out.
// ---------------------------------------------------------------------------
__global__ void __launch_bounds__(256)
batch_mean_wmma_kernel(const float* __restrict__ E_ws, float* __restrict__ out) {
    const int gtid = blockIdx.x * blockDim.x + threadIdx.x;
    const int wave = gtid >> 5;                // 4096 waves total
    const int lane = threadIdx.x & 31;
    const int pbase = wave * 16;               // 16 outputs per wave
    const int r  = lane & 15;                  // A-layout: lane = M row
    const int kk = (lane >> 4) << 1;           // lanes 0-15 -> K{0,1}; 16-31 -> K{2,3}

    v8f acc = {};
    v2f bmat; bmat.x = 1.0f / 64.0f; bmat.y = 1.0f / 64.0f;   // all B[k][c] equal

    for (int kb = 0; kb < BATCH; kb += 4) {
        v2f a;
        a.x = E_ws[(size_t)(kb + kk)     * (M * N) + pbase + r];
        a.y = E_ws[(size_t)(kb + kk + 1) * (M * N) + pbase + r];
        // 8 args: (neg_a, A, neg_b, B, c_mod, C, reuse_a, reuse_b)
        acc = __builtin_amdgcn_wmma_f32_16x16x4_f32(
            false, a, false, bmat, (short)0, acc, false, false);
    }
    // D layout: lane 0 holds D[M=0..7][N=0] in acc[0..7]; lane 16 holds M=8..15.
    if (lane == 0) {
        #pragma unroll
        for (int v = 0; v < 8; ++v) out[pbase + v] = acc[v];
    } else if (lane == 16) {
        #pragma unroll
        for (int v = 0; v < 8; ++v) out[pbase + 8 + v] = acc[v];
    }
}

extern "C" void kernel_launch(void* const* d_in, const int* in_sizes, int n_in,
                              void* d_out, int out_size, void* d_ws, size_t ws_size,
                              hipStream_t stream) {
    (void)in_sizes; (void)n_in; (void)out_size; (void)ws_size;
    const float* D = (const float*)d_in[0];        // [64, 256, 256] fp32
    float* out = (float*)d_out;                    // [256, 256] fp32

    // Workspace layout: R (64*256*256 f32) then E (64*256*256 f32) = 33.6 MB.
    float* R_ws = (float*)d_ws;
    float* E_ws = R_ws + (size_t)BATCH * M * N;

    dtw_grad_kernel<<<BATCH, 256, 0, stream>>>(D, R_ws, E_ws);

    // 65536 outputs / 16 per wave = 4096 waves = 512 blocks of 256 threads.
    batch_mean_wmma_kernel<<<(M * N) / 16 / 8, 256, 0, stream>>>(E_ws, out);
}